// NeuSRenderer_89386859365072
// MI455X (gfx1250) — compile-verified
//
#include <hip/hip_runtime.h>
#include <stdint.h>

// NeuS importance sampling (z_samples) for MI455X / gfx1250.
//
// Bandwidth-bound kernel (~100 MB traffic, ~0.3 GFLOP): the CDNA5 feature
// that matters is the Tensor Data Mover. Each block DMAs its z_vals and sdf
// tiles (64 rays x 64 f32) global->LDS with ONE tensor_load_to_lds each,
// using the TDM's hardware LDS padding (1 dword pad per 64-dword row) to get
// a 65-dword row stride => bank-conflict-free per-ray row access in LDS.
// Per-ray global loads are issued BEFORE the tensor-wait/barrier so they
// overlap the DMA.

#define N_RAYS   131072
#define NS       64      // samples per ray (bins)
#define NI       64      // importance samples per ray (output)
#define ROWS     64      // rays per block (one thread per ray)
#define STRIDE   65      // padded LDS row stride in floats (64 + TDM pad)

typedef unsigned int u32x4 __attribute__((ext_vector_type(4)));
typedef int          i32x4 __attribute__((ext_vector_type(4)));
typedef int          i32x8 __attribute__((ext_vector_type(8)));

// Issue one TDM load of a 2-D tile: ROWS rows x NS f32, contiguous in global
// memory, written to LDS with a 1-dword pad after every 64 dwords.
// Descriptor layout per CDNA5 ISA §8.3 (group 0) / §8.4 (group 1).
__device__ __forceinline__ void tdm_load_tile(const float* gbase, uint32_t lds_byte_off)
{
    uint64_t ga = (uint64_t)(uintptr_t)gbase;

    u32x4 g0;
    g0.x = 1u;                                   // count=1 (valid), user mode
    g0.y = lds_byte_off;                         // lds_addr (bytes)
    g0.z = (uint32_t)(ga & 0xffffffffu);         // global_addr[31:0]
    g0.w = (uint32_t)((ga >> 32) & 0x01ffffffu)  // global_addr[56:32]
         | (2u << 30);                           // type = 2 ("image")

    i32x8 g1;
    g1[0] = (2 << 16)                            // data_size = 4 bytes
          | (1 << 20)                            // pad_enable
          | (5 << 22);                           // pad_interval: 2^(5+1)=64 dwords
                                                 // pad_amount = 0 => 1 dword
    g1[1] = (NS   & 0xffff) << 16;               // tensor_dim0 = 64
    g1[2] = (ROWS & 0xffff) << 16;               // tensor_dim1 = 64 (rows in tile)
    g1[3] = (NS   & 0xffff) << 16;               // tile_dim0 = 64
    g1[4] = ROWS;                                // tile_dim1 = 64, tile_dim2 = 0
    g1[5] = NS;                                  // tensor_dim0_stride = 64 elements
    g1[6] = 0;                                   // dim0_stride hi / dim1_stride lo
    g1[7] = 0;                                   // dim1_stride hi

    i32x4 gz = {0, 0, 0, 0};                     // groups 2/3 unused (2-D tensor)
#if defined(__clang_major__) && (__clang_major__ >= 23)
    i32x8 gz8 = {0, 0, 0, 0, 0, 0, 0, 0};
    __builtin_amdgcn_tensor_load_to_lds(g0, g1, gz, gz, gz8, 0);
#else
    __builtin_amdgcn_tensor_load_to_lds(g0, g1, gz, gz, 0);
#endif
}

__global__ __launch_bounds__(ROWS) void neus_zsample_kernel(
    const float* __restrict__ rays_o, const float* __restrict__ rays_d,
    const float* __restrict__ z_vals, const float* __restrict__ sdf,
    const int* __restrict__ inv_s_p, float* __restrict__ out)
{
    __shared__ float zl[ROWS * STRIDE];   // z_vals tile (padded rows)
    __shared__ float sl[ROWS * STRIDE];   // sdf tile; later reused as out stage
    __shared__ float cl[ROWS * STRIDE];   // per-ray unnormalized CDF

    const int t    = threadIdx.x;
    const int ray0 = blockIdx.x * ROWS;

    // Wave 0 only: TDM ignores EXEC and issues once per executing wave, so a
    // single wave must issue the DMA to avoid duplicated traffic.
    if (t < 32) {
        tdm_load_tile(z_vals + (size_t)ray0 * NS, (uint32_t)(uintptr_t)&zl[0]);
        tdm_load_tile(sdf    + (size_t)ray0 * NS, (uint32_t)(uintptr_t)&sl[0]);
    }

    // Per-ray loads issued now so they overlap the in-flight tensor DMA.
    const int   ray   = ray0 + t;
    const float ox = rays_o[ray * 3 + 0], oy = rays_o[ray * 3 + 1], oz = rays_o[ray * 3 + 2];
    const float dx = rays_d[ray * 3 + 0], dy = rays_d[ray * 3 + 1], dzc = rays_d[ray * 3 + 2];
    const float inv_s = (float)inv_s_p[0];

    // |o + d z|^2 = |o|^2 + 2(o.d) z + z^2   (rays_d is unit length)
    const float aq = ox * ox + oy * oy + oz * oz;
    const float bq = 2.0f * (ox * dx + oy * dy + oz * dzc);

    if (t < 32) {
        __builtin_amdgcn_s_wait_tensorcnt(0);
    }
    __syncthreads();

    const float* zr = &zl[t * STRIDE];
    const float* sr = &sl[t * STRIDE];
    float*       cr = &cl[t * STRIDE];

    // ---- Pass 1: alpha compositing scan -> unnormalized CDF in cr[0..63] ----
    float z0 = zr[0], s0 = sr[0];
    bool  in0 = fmaf(z0, z0 + bq, aq) < 1.0f;     // radius^2 < 1
    float prev_raw = 0.0f;                        // raw cos_val[i-1], cos_val[-1]=0
    float T = 1.0f;                               // transmittance (cumprod)
    float csum = 0.0f;                            // running sum of (weights+1e-5)
    cr[0] = 0.0f;
    for (int i = 0; i < NS - 1; ++i) {
        float z1 = zr[i + 1], s1 = sr[i + 1];
        bool  in1 = fmaf(z1, z1 + bq, aq) < 1.0f;
        float inside = (in0 | in1) ? 1.0f : 0.0f;
        float dz  = z1 - z0;
        float mid = 0.5f * (s0 + s1);
        float raw = (s1 - s0) / (dz + 1e-5f);
        float cv  = fminf(prev_raw, raw);
        cv = fminf(fmaxf(cv, -1000.0f), 0.0f) * inside;
        prev_raw = raw;
        float h  = cv * dz * 0.5f;
        float pc = 1.0f / (1.0f + __expf(-(mid - h) * inv_s));  // sigmoid
        float nc = 1.0f / (1.0f + __expf(-(mid + h) * inv_s));
        float alpha = (pc - nc + 1e-5f) / (pc + 1e-5f);
        float w = alpha * T + 1e-5f;              // weights + 1e-5 (sample_pdf)
        T *= (1.0f - alpha + 1e-7f);
        csum += w;
        cr[i + 1] = csum;                         // unnormalized cumsum
        z0 = z1; s0 = s1; in0 = in1;
    }
    const float rnorm = 1.0f / csum;

    __syncthreads();   // all pass-1 reads of sl done before reusing it below

    // ---- Pass 2: inverse-CDF sampling (two-pointer merge; u and cdf sorted) ----
    int pos = 1;                                  // cr[0]=0 < u*csum always
    for (int j = 0; j < NI; ++j) {
        float u  = ((float)j + 0.5f) * (1.0f / (float)NI);
        float ut = u * csum;                      // compare unnormalized
        while (pos < NS && cr[pos] <= ut) ++pos;  // searchsorted(..., 'right')
        int below = pos - 1;
        int above = (pos < NS) ? pos : NS - 1;
        float cb = cr[below] * rnorm;
        float ca = cr[above] * rnorm;
        float denom = ca - cb;
        denom = (denom < 1e-5f) ? 1.0f : denom;
        float tt = (u - cb) / denom;
        float zb = zr[below], za = zr[above];
        sl[t * STRIDE + j] = zb + tt * (za - zb); // stage for coalesced store
    }
    __syncthreads();

    // ---- Coalesced copy-out of the staged tile ----
    float* outb = out + (size_t)ray0 * NI;
    for (int c = t; c < ROWS * NI; c += ROWS) {
        int row = c >> 6;                         // c / 64
        int col = c & 63;
        outb[c] = sl[row * STRIDE + col];
    }
}

extern "C" void kernel_launch(void* const* d_in, const int* in_sizes, int n_in,
                              void* d_out, int out_size, void* d_ws, size_t ws_size,
                              hipStream_t stream) {
    (void)in_sizes; (void)n_in; (void)out_size; (void)d_ws; (void)ws_size;
    const float* rays_o = (const float*)d_in[0];
    const float* rays_d = (const float*)d_in[1];
    const float* z_vals = (const float*)d_in[2];
    const float* sdf    = (const float*)d_in[3];
    // d_in[4] = n_importance (==64, compile-time), d_in[5] = inv_s (int scalar)
    const int*   inv_s  = (const int*)d_in[5];
    float* out = (float*)d_out;

    const int nblocks = N_RAYS / ROWS;   // 2048
    hipLaunchKernelGGL(neus_zsample_kernel, dim3(nblocks), dim3(ROWS), 0, stream,
                       rays_o, rays_d, z_vals, sdf, inv_s, out);
}